// Model_38912403702170
// MI455X (gfx1250) — compile-verified
//
#include <hip/hip_runtime.h>
#include <math.h>

#define LSEQ 50
#define DIM 64
#define NSESS 1024
#define NITEMS 100000
#define NT 5   // 5 x 16 = 80 item columns per wave; 100000 / 80 = 1250 strips

typedef float v2f __attribute__((ext_vector_type(2)));
typedef float v8f __attribute__((ext_vector_type(8)));

// ---------------------------------------------------------------------------
// Phase A: one 64-thread block per session. Builds the session graph, runs the
// gated-GRU step and attention readout, writes s[b][0:64] into workspace S.
// All small matmuls are dim-parallel (thread d owns output dim d) over LDS.
// ---------------------------------------------------------------------------
__global__ __launch_bounds__(64) void session_kernel(
    const int* __restrict__ x,
    const float* __restrict__ emb,
    const float* __restrict__ w_in,
    const float* __restrict__ w_out,
    const float* __restrict__ w_ih,   // (192,128)
    const float* __restrict__ w_hh,   // (192,64)
    const float* __restrict__ b_ih,   // (192)
    const float* __restrict__ b_hh,   // (192)
    const float* __restrict__ w_read, // (64,64)
    float* __restrict__ S)            // (1024,64)
{
    __shared__ int   s_comp[LSEQ];
    __shared__ int   s_uniq[LSEQ];
    __shared__ int   s_inv[LSEQ];
    __shared__ int   s_n, s_nn;
    __shared__ float s_ne [LSEQ * DIM];   // node embeddings
    __shared__ float s_t  [LSEQ * DIM];   // Hin -> Hout -> h (reused)
    __shared__ float s_ain[LSEQ * DIM];
    __shared__ float s_aout[LSEQ * DIM];
    __shared__ float s_q[DIM];
    __shared__ float s_att[LSEQ];

    const int b = blockIdx.x;
    const int d = threadIdx.x;   // output dimension owned by this thread

    if (d == 0) {
        // stable compaction of non-zero items (matches argsort(~valid))
        int n = 0;
        for (int i = 0; i < LSEQ; ++i) {
            int v = x[b * LSEQ + i];
            if (v != 0) s_comp[n++] = v;
        }
        // run-length node assignment (change = first or differs from prev)
        int nn = 0;
        for (int i = 0; i < n; ++i) {
            if (i == 0 || s_comp[i] != s_comp[i - 1]) { s_uniq[nn] = s_comp[i]; ++nn; }
            s_inv[i] = nn - 1;
        }
        s_n = n; s_nn = nn;
    }
    __syncthreads();
    const int n  = s_n;
    const int nn = s_nn;

    if (n == 0) {            // empty session -> scores forced to 0 via s = 0
        S[b * DIM + d] = 0.0f;
        return;
    }

    // node embeddings + zero aggregators
    for (int j = 0; j < nn; ++j) {
        s_ne [j * DIM + d] = emb[(long)s_uniq[j] * DIM + d];
        s_ain[j * DIM + d] = 0.0f;
        s_aout[j * DIM + d] = 0.0f;
    }
    __syncthreads();

    // Hin = node_emb @ w_in.T ; agg_in[dst] += Hin[src]
    {
        const float* wr = w_in + d * DIM;
        for (int j = 0; j < nn; ++j) {
            float acc = 0.0f;
            for (int k = 0; k < DIM; ++k) acc += s_ne[j * DIM + k] * wr[k];
            s_t[j * DIM + d] = acc;
        }
    }
    __syncthreads();
    for (int i = 0; i + 1 < n; ++i)
        s_ain[s_inv[i + 1] * DIM + d] += s_t[s_inv[i] * DIM + d];
    __syncthreads();

    // Hout = node_emb @ w_out.T ; agg_out[src] += Hout[dst]
    {
        const float* wr = w_out + d * DIM;
        for (int j = 0; j < nn; ++j) {
            float acc = 0.0f;
            for (int k = 0; k < DIM; ++k) acc += s_ne[j * DIM + k] * wr[k];
            s_t[j * DIM + d] = acc;
        }
    }
    __syncthreads();
    for (int i = 0; i + 1 < n; ++i)
        s_aout[s_inv[i] * DIM + d] += s_t[s_inv[i + 1] * DIM + d];
    __syncthreads();

    // GRU cell; h written into s_t
    {
        const float bi_r = b_ih[d], bi_z = b_ih[DIM + d], bi_n = b_ih[2 * DIM + d];
        const float bh_r = b_hh[d], bh_z = b_hh[DIM + d], bh_n = b_hh[2 * DIM + d];
        const float* wir = w_ih + (0 * DIM + d) * (2 * DIM);
        const float* wiz = w_ih + (1 * DIM + d) * (2 * DIM);
        const float* win = w_ih + (2 * DIM + d) * (2 * DIM);
        const float* whr = w_hh + (0 * DIM + d) * DIM;
        const float* whz = w_hh + (1 * DIM + d) * DIM;
        const float* whn = w_hh + (2 * DIM + d) * DIM;
        for (int j = 0; j < nn; ++j) {
            float ir = bi_r, iz = bi_z, inn = bi_n;
            for (int k = 0; k < DIM; ++k) {
                float a0 = s_ain[j * DIM + k];
                float a1 = s_aout[j * DIM + k];
                ir  += a0 * wir[k] + a1 * wir[DIM + k];
                iz  += a0 * wiz[k] + a1 * wiz[DIM + k];
                inn += a0 * win[k] + a1 * win[DIM + k];
            }
            float hr = bh_r, hz = bh_z, hn = bh_n;
            for (int k = 0; k < DIM; ++k) {
                float ne = s_ne[j * DIM + k];
                hr += ne * whr[k];
                hz += ne * whz[k];
                hn += ne * whn[k];
            }
            float r  = 1.0f / (1.0f + expf(-(ir + hr)));
            float z  = 1.0f / (1.0f + expf(-(iz + hz)));
            float ng = tanhf(inn + r * hn);
            float ne_d = s_ne[j * DIM + d];
            s_t[j * DIM + d] = (n >= 2) ? ((1.0f - z) * ng + z * ne_d) : ne_d;
        }
    }
    __syncthreads();

    // q = h[last] @ w_read.T
    {
        const int last = s_inv[n - 1];
        const float* wr = w_read + d * DIM;
        float acc = 0.0f;
        for (int k = 0; k < DIM; ++k) acc += s_t[last * DIM + k] * wr[k];
        s_q[d] = acc;
    }
    __syncthreads();

    // attention logits (only valid nodes matter; masked ones get exp(-1e9)=0)
    if (d < nn) {
        float acc = 0.0f;
        for (int k = 0; k < DIM; ++k) acc += s_t[d * DIM + k] * s_q[k];
        s_att[d] = acc;
    }
    __syncthreads();
    if (d == 0) {
        float m = -1e30f;
        for (int j = 0; j < nn; ++j) m = fmaxf(m, s_att[j]);
        float sum = 0.0f;
        for (int j = 0; j < nn; ++j) { float e = expf(s_att[j] - m); s_att[j] = e; sum += e; }
        float inv = 1.0f / sum;
        for (int j = 0; j < nn; ++j) s_att[j] *= inv;
    }
    __syncthreads();

    // s = att @ h
    {
        float acc = 0.0f;
        for (int j = 0; j < nn; ++j) acc += s_att[j] * s_t[j * DIM + d];
        S[b * DIM + d] = acc;
    }
}

// ---------------------------------------------------------------------------
// Phase B: scores = S (1024x64) @ emb^T (64x100000) using V_WMMA_F32_16X16X4.
// Each wave computes a 16(M) x 80(N) strip: 5 accumulators, 16 K-steps.
// Lane layout per ISA 7.12.2: lo = lane&15 selects M (A) / N (B) row,
// hi = lane>>4 plus vgpr index select K -> contiguous float2 loads.
// Output streamed with non-temporal stores (keep emb resident in L2).
// ---------------------------------------------------------------------------
__global__ __launch_bounds__(256) void score_kernel(
    const float* __restrict__ S,
    const float* __restrict__ emb,
    float* __restrict__ out)
{
    const int wave = threadIdx.x >> 5;
    const int lane = threadIdx.x & 31;
    const int lo   = lane & 15;
    const int hi   = lane >> 4;

    const int tile = blockIdx.x * 8 + wave;  // 0 .. 79999
    const int mt   = tile & 63;              // 64 M-tiles, varies fastest:
    const int ns   = tile >> 6;              // all 8 waves in a block share ns
    const int m0   = mt * 16;
    const int n0   = ns * (16 * NT);

    const float* aRow = S   + (size_t)(m0 + lo) * DIM + 2 * hi;
    const float* bRow = emb + (size_t)(n0 + lo) * DIM + 2 * hi;

    v8f acc[NT] = {};

#pragma unroll
    for (int k0 = 0; k0 < DIM; k0 += 4) {
        v2f a = *(const v2f*)(aRow + k0);
#pragma unroll
        for (int t = 0; t < NT; ++t) {
            v2f bm = *(const v2f*)(bRow + (size_t)t * 16 * DIM + k0);
            acc[t] = __builtin_amdgcn_wmma_f32_16x16x4_f32(
                false, a, false, bm, (short)0, acc[t], false, false);
        }
    }

#pragma unroll
    for (int t = 0; t < NT; ++t) {
#pragma unroll
        for (int v = 0; v < 8; ++v) {
            size_t row = (size_t)(m0 + v + 8 * hi);
            __builtin_nontemporal_store(acc[t][v],
                out + row * (size_t)NITEMS + n0 + t * 16 + lo);
        }
    }
}

// ---------------------------------------------------------------------------
extern "C" void kernel_launch(void* const* d_in, const int* in_sizes, int n_in,
                              void* d_out, int out_size, void* d_ws, size_t ws_size,
                              hipStream_t stream) {
    const int*   x      = (const int*)  d_in[0];
    // d_in[1] = attn_mask: unused by the reference computation
    const float* emb    = (const float*)d_in[2];
    const float* w_in   = (const float*)d_in[3];
    const float* w_out  = (const float*)d_in[4];
    const float* w_ih   = (const float*)d_in[5];
    const float* w_hh   = (const float*)d_in[6];
    const float* b_ih   = (const float*)d_in[7];
    const float* b_hh   = (const float*)d_in[8];
    const float* w_read = (const float*)d_in[9];

    float* out = (float*)d_out;
    float* S   = (float*)d_ws;           // 1024 x 64 session vectors

    session_kernel<<<NSESS, 64, 0, stream>>>(x, emb, w_in, w_out, w_ih, w_hh,
                                             b_ih, b_hh, w_read, S);

    // 64 M-tiles * 1250 N-strips = 80000 waves -> 10000 blocks of 8 waves
    score_kernel<<<10000, 256, 0, stream>>>(S, emb, out);
}